// mLSTMIntraLayer_69355131896513
// MI455X (gfx1250) — compile-verified
//
#include <hip/hip_runtime.h>

typedef __attribute__((ext_vector_type(2))) float v2f;
typedef __attribute__((ext_vector_type(8))) float v8f;
typedef __attribute__((ext_vector_type(4))) unsigned int v4u;
typedef __attribute__((ext_vector_type(8))) int v8i;
typedef __attribute__((ext_vector_type(4))) int v4i;

#define LDSS 68   // padded LDS row stride (floats): conflict-free, 16B-aligned

// Issue one TDM tensor load: 64x64 f32 tile, row-major in memory (stride 64),
// written to LDS with 4 DWORDs of padding after every 64 DWORDs (stride 68).
__device__ __forceinline__ void tdm_load_tile_64x64(unsigned lds_off,
                                                    const float* gptr) {
    unsigned long long ga = (unsigned long long)(uintptr_t)gptr;
    v4u g0;
    g0[0] = 1u;                                   // count=1, user descriptor
    g0[1] = lds_off;                              // lds_addr (bytes)
    g0[2] = (unsigned)(ga & 0xFFFFFFFFu);         // global_addr[31:0]
    g0[3] = (unsigned)((ga >> 32) & 0x01FFFFFFu)  // global_addr[56:32]
          | (2u << 30);                           // type=2 ("image")
    v8i g1;
    g1[0] = (2 << 16)      // data_size = 2 -> 4 bytes
          | (1 << 20)      // pad_enable
          | (5 << 22)      // pad_interval = 5 -> every 64 DWORDs
          | (3 << 25);     // pad_amount  = 3 -> 4 DWORDs
    g1[1] = (64 << 16);    // tensor_dim0[15:0] = 64 (bits 63:48)
    g1[2] = (64 << 16);    // tensor_dim0[31:16]=0 ; tensor_dim1[15:0] = 64
    g1[3] = (64 << 16);    // tensor_dim1[31:16]=0 ; tile_dim0 = 64
    g1[4] = 64;            // tile_dim1 = 64 ; tile_dim2 = 0
    g1[5] = 64;            // tensor_dim0_stride[31:0] = 64 elements
    g1[6] = 0;             // stride0[47:32]=0 ; tensor_dim1_stride[15:0]=0
    g1[7] = 0;
    v4i gz4 = {0, 0, 0, 0};
    v8i gz8 = {0, 0, 0, 0, 0, 0, 0, 0};
    __builtin_amdgcn_tensor_load_to_lds(g0, g1, gz4, gz4, gz8, 0);
}

__global__ __launch_bounds__(128) void mlstm_intra_kernel(
    const float* __restrict__ q, const float* __restrict__ k,
    const float* __restrict__ v, const float* __restrict__ ig,
    const float* __restrict__ fg, float* __restrict__ out)
{
    __shared__ float Qs[64 * LDSS];   // Q tile, later aliased as C = scores*D
    __shared__ float Ks[64 * LDSS];
    __shared__ float Vs[64 * LDSS];
    __shared__ float g_s[64];         // g[j]  = ig[j] - cum[j]
    __shared__ float r_s[64];         // r[i]  = prefix max of g
    __shared__ float m_s[64];         // m[i]  = cum[i] + r[i]
    __shared__ float inv_s[64];       // 1 / (n[i] + eps)

    const int blk = blockIdx.x;
    const int bh  = blk >> 6;         // batch*head index
    const int ch  = blk & 63;         // chunk index
    const size_t gbase = (size_t)bh * 4096 + (size_t)ch * 64;  // gate offset
    const size_t base  = gbase * 64;                            // q/k/v/out offset

    const int tid = threadIdx.x;

    // ---- Phase 0: TDM DMA of Q,K,V chunks into padded LDS (wave 0 only) ---
    if (tid < 32) {
        if (tid == 0) {
            tdm_load_tile_64x64((unsigned)(uintptr_t)Qs, q + base);
            tdm_load_tile_64x64((unsigned)(uintptr_t)Ks, k + base);
            tdm_load_tile_64x64((unsigned)(uintptr_t)Vs, v + base);

            // gate scan overlaps the DMA: cum-logsigmoid, g, prefix max, m
            float cum = 0.f, rmax = -3.4e38f;
            for (int j = 0; j < 64; ++j) {
                float f  = fg[gbase + j];
                float lf = fminf(f, 0.f) - log1pf(__expf(-fabsf(f)));
                cum += lf;
                float gj = ig[gbase + j] - cum;
                rmax = fmaxf(rmax, gj);
                g_s[j] = gj;
                r_s[j] = rmax;
                m_s[j] = cum + rmax;
            }
        }
        __builtin_amdgcn_s_wait_tensorcnt(0);   // wave 0: DMA complete
    }
    __syncthreads();                            // publish LDS to all waves

    const int wave = tid >> 5;
    const int lane = tid & 31;
    const int half = lane >> 4;       // K-group / row-half selector
    const int lm   = lane & 15;
    const int rowA = wave * 16 + lm;  // A-fragment row for this wave
    const int kb   = half << 1;       // K base within group of 4

    // ---- Phase 1: scores = (Q K^T)/8, gated by D, stored back to LDS -----
    v8f acc[4] = {};
    for (int kk = 0; kk < 16; ++kk) {
        v2f a;
        a.x = Qs[rowA * LDSS + (kk << 2) + kb];
        a.y = Qs[rowA * LDSS + (kk << 2) + kb + 1];
#pragma unroll
        for (int tj = 0; tj < 4; ++tj) {
            v2f b;
            b.x = Ks[(tj * 16 + lm) * LDSS + (kk << 2) + kb];
            b.y = Ks[(tj * 16 + lm) * LDSS + (kk << 2) + kb + 1];
            acc[tj] = __builtin_amdgcn_wmma_f32_16x16x4_f32(
                false, a, false, b, (short)0, acc[tj], false, false);
        }
    }

    float ri[8];
#pragma unroll
    for (int r = 0; r < 8; ++r) ri[r] = r_s[wave * 16 + r + 8 * half];

#pragma unroll
    for (int tj = 0; tj < 4; ++tj) {
        int   j  = tj * 16 + lm;
        float gj = g_s[j];
#pragma unroll
        for (int r = 0; r < 8; ++r) {
            int   i = wave * 16 + r + 8 * half;
            float d = (j <= i) ? __expf(gj - ri[r]) : 0.f;   // decay mask
            Qs[i * LDSS + j] = acc[tj][r] * 0.125f * d;      // C aliases Qs
        }
    }
    __syncthreads();

    // ---- Phase 2: stabilized normalizer per row ---------------------------
    if (tid < 64) {
        float s = 0.f;
        for (int j = 0; j < 64; ++j) s += Qs[tid * LDSS + j];
        float n = fmaxf(fabsf(s), __expf(-m_s[tid]));
        inv_s[tid] = 1.0f / (n + 1e-6f);
    }
    __syncthreads();

    // ---- Phase 3: H = C V, scaled by inv[i], written out ------------------
    v8f h[4] = {};
    for (int kk = 0; kk < 16; ++kk) {
        v2f a;
        a.x = Qs[rowA * LDSS + (kk << 2) + kb];
        a.y = Qs[rowA * LDSS + (kk << 2) + kb + 1];
#pragma unroll
        for (int tj = 0; tj < 4; ++tj) {
            int kr = (kk << 2) + kb;
            v2f b;
            b.x = Vs[kr * LDSS + tj * 16 + lm];
            b.y = Vs[(kr + 1) * LDSS + tj * 16 + lm];
            h[tj] = __builtin_amdgcn_wmma_f32_16x16x4_f32(
                false, a, false, b, (short)0, h[tj], false, false);
        }
    }

    float iv[8];
#pragma unroll
    for (int r = 0; r < 8; ++r) iv[r] = inv_s[wave * 16 + r + 8 * half];

#pragma unroll
    for (int tj = 0; tj < 4; ++tj) {
#pragma unroll
        for (int r = 0; r < 8; ++r) {
            int i = wave * 16 + r + 8 * half;
            out[base + (size_t)i * 64 + tj * 16 + lm] = h[tj][r] * iv[r];
        }
    }
}

extern "C" void kernel_launch(void* const* d_in, const int* in_sizes, int n_in,
                              void* d_out, int out_size, void* d_ws, size_t ws_size,
                              hipStream_t stream) {
    const float* q  = (const float*)d_in[0];
    const float* k  = (const float*)d_in[1];
    const float* v  = (const float*)d_in[2];
    const float* ig = (const float*)d_in[3];
    const float* fg = (const float*)d_in[4];
    float* out = (float*)d_out;

    // q is [B,NH,S=4096,DH=64]; one workgroup per (b,nh,chunk of 64)
    const int nbh  = in_sizes[0] / (4096 * 64);  // B*NH
    const int grid = nbh * (4096 / 64);          // B*NH*NC

    mlstm_intra_kernel<<<grid, 128, 0, stream>>>(q, k, v, ig, fg, out);
}